// SparseGapModel_1460288881187
// MI455X (gfx1250) — compile-verified
//
#include <hip/hip_runtime.h>

typedef __attribute__((ext_vector_type(2))) float v2f;
typedef __attribute__((ext_vector_type(8))) float v8f;

#define FEAT 128
#define NSUP 256
#define NSPEC 4
#define QS_STRIDE 132   // floats; 132*4 bytes row pitch -> conflict-free LDS banks

// ---------------------------------------------------------------------------
// Kernel 1: A_s = sp[s]^T diag(w[s]) sp[s]   -> 4 matrices of 128x128 f32
// grid: 4*128 blocks, 128 threads.  block = (s, row a), thread = col b.
// ---------------------------------------------------------------------------
__global__ void __launch_bounds__(128)
k_build_A(const float* __restrict__ sp, const float* __restrict__ w,
          float* __restrict__ A) {
    int s = blockIdx.x >> 7;
    int a = blockIdx.x & 127;
    int b = threadIdx.x;
    const float* sps = sp + (size_t)s * NSUP * FEAT;
    const float* ws  = w + s * NSUP;
    float acc = 0.f;
    for (int m = 0; m < NSUP; ++m) {
        float va = sps[m * FEAT + a];     // uniform across threads
        float vb = sps[m * FEAT + b];     // coalesced
        acc = fmaf(ws[m] * va, vb, acc);
    }
    A[((size_t)s * FEAT + a) * FEAT + b] = acc;
}

// ---------------------------------------------------------------------------
// Binning: counts -> padded prefix -> scatter atom indices per species.
// ---------------------------------------------------------------------------
__global__ void k_init(int* counts, int* cursor, int* idxlist, int n_pad) {
    int t = blockIdx.x * blockDim.x + threadIdx.x;
    if (t < NSPEC) { counts[t] = 0; cursor[t] = 0; }
    if (t < n_pad) idxlist[t] = -1;
}

__global__ void k_count(const int* __restrict__ species, int n, int* counts) {
    int t = blockIdx.x * blockDim.x + threadIdx.x;
    if (t < n) atomicAdd(&counts[species[t] & 3], 1);
}

__global__ void k_scan(const int* __restrict__ counts, int* base, int* tileOff) {
    if (blockIdx.x == 0 && threadIdx.x == 0) {
        int b = 0;
        for (int s = 0; s < NSPEC; ++s) {
            base[s]    = b;
            tileOff[s] = b >> 4;
            b += (counts[s] + 15) & ~15;   // pad each bin to multiple of 16
        }
        base[NSPEC]    = b;
        tileOff[NSPEC] = b >> 4;
    }
}

__global__ void k_scatter(const int* __restrict__ species, int n,
                          const int* __restrict__ base, int* cursor,
                          int* __restrict__ idxlist) {
    int t = blockIdx.x * blockDim.x + threadIdx.x;
    if (t < n) {
        int s = species[t] & 3;
        int p = atomicAdd(&cursor[s], 1);
        idxlist[base[s] + p] = t;
    }
}

// ---------------------------------------------------------------------------
// Main kernel: one wave (32 lanes) per 16-atom tile of a single species.
//   q = normalized rows (16x128 in LDS)
//   Y = Q @ A_s  via 8 col-tiles x 32 k-steps of V_WMMA_F32_16X16X4_F32
//   e_m = sum_c Y[m,c] * Q[m,c]  -> per_atom[orig index]
// ---------------------------------------------------------------------------
__global__ void __launch_bounds__(32)
k_main(const float* __restrict__ ps, const float* __restrict__ Aall,
       const int* __restrict__ idxlist, const int* __restrict__ base,
       const int* __restrict__ tileOff, float* __restrict__ per_atom) {
    __shared__ float qs[16 * QS_STRIDE];
    __shared__ int   sidx[16];
    __shared__ float red[2 * 8 * 16];

    int t  = blockIdx.x;
    if (t >= tileOff[NSPEC]) return;          // uniform exit, EXEC stays full
    int s = 0;
    if (t >= tileOff[1]) s = 1;
    if (t >= tileOff[2]) s = 2;
    if (t >= tileOff[3]) s = 3;
    int abase = base[s] + (t - tileOff[s]) * 16;

    int l   = threadIdx.x;                    // 0..31
    int col = l & 15;
    int hi  = l >> 4;

    if (l < 16) sidx[l] = idxlist[abase + l];
    __syncthreads();

    // ---- load 16 feature rows (float4 per lane per row) ----
    for (int r = 0; r < 16; ++r) {
        int ai = sidx[r];
        float4 v = make_float4(0.f, 0.f, 0.f, 0.f);
        if (ai >= 0) v = *(const float4*)(ps + (size_t)ai * FEAT + l * 4);
        *(float4*)(&qs[r * QS_STRIDE + l * 4]) = v;
    }
    __syncthreads();

    // ---- normalize rows: lane handles row (l&15), half (hi) of columns ----
    {
        float* q = qs + col * QS_STRIDE + hi * 64;
        float ss = 0.f;
        #pragma unroll 8
        for (int k = 0; k < 64; ++k) ss = fmaf(q[k], q[k], ss);
        ss += __shfl_xor(ss, 16, 32);
        float inv = (ss > 0.f) ? (1.0f / sqrtf(ss)) : 0.f;
        #pragma unroll 8
        for (int k = 0; k < 64; ++k) q[k] *= inv;
    }
    __syncthreads();

    // ---- Y = Q @ A_s via f32 WMMA ----
    const float* Amat = Aall + (size_t)s * FEAT * FEAT;
    const float* qrow = qs + col * QS_STRIDE;   // A-frag row = l&15
    int koff = hi * 2;                          // A and B share K sub-offset

    v8f acc[8];
    #pragma unroll
    for (int n = 0; n < 8; ++n) acc[n] = (v8f){0.f,0.f,0.f,0.f,0.f,0.f,0.f,0.f};

    #pragma unroll 2
    for (int kk = 0; kk < 32; ++kk) {
        int kb = kk * 4;
        v2f a;
        a.x = qrow[kb + koff];
        a.y = qrow[kb + koff + 1];
        const float* B0 = Amat + (size_t)(kb + koff) * FEAT + col;
        const float* B1 = B0 + FEAT;
        #pragma unroll
        for (int n = 0; n < 8; ++n) {
            v2f b;
            b.x = B0[n * 16];
            b.y = B1[n * 16];
            acc[n] = __builtin_amdgcn_wmma_f32_16x16x4_f32(
                false, a, false, b, (short)0, acc[n], false, false);
        }
    }

    // ---- e_m = sum_c Y[m,c]*Q[m,c]; lane holds row v+8*hi, col 16n+col ----
    #pragma unroll
    for (int v = 0; v < 8; ++v) {
        float p = 0.f;
        #pragma unroll
        for (int n = 0; n < 8; ++n)
            p = fmaf(acc[n][v], qs[(v + 8 * hi) * QS_STRIDE + n * 16 + col], p);
        red[(hi * 8 + v) * 16 + col] = p;
    }
    __syncthreads();

    if (l < 16) {
        int m = l;
        float e = 0.f;
        #pragma unroll
        for (int c = 0; c < 16; ++c) e += red[((m >> 3) * 8 + (m & 7)) * 16 + c];
        int ai = sidx[m];
        if (ai >= 0) per_atom[ai] = e;
    }
}

// ---------------------------------------------------------------------------
// Segment sum over sorted structure ids (deterministic, no float atomics).
// One block per structure; binary search for its contiguous atom range.
// ---------------------------------------------------------------------------
__global__ void __launch_bounds__(256)
k_segsum(const float* __restrict__ per_atom, const int* __restrict__ sid,
         int n, float* __restrict__ out) {
    __shared__ float sred[256];
    __shared__ int range[2];
    int b = blockIdx.x;
    if (threadIdx.x == 0) {
        int lo = 0, h = n;
        while (lo < h) { int mid = (lo + h) >> 1; if (sid[mid] < b) lo = mid + 1; else h = mid; }
        range[0] = lo;
        int lo2 = lo, h2 = n;
        while (lo2 < h2) { int mid = (lo2 + h2) >> 1; if (sid[mid] < b + 1) lo2 = mid + 1; else h2 = mid; }
        range[1] = lo2;
    }
    __syncthreads();
    float acc = 0.f;
    for (int i = range[0] + threadIdx.x; i < range[1]; i += 256) acc += per_atom[i];
    sred[threadIdx.x] = acc;
    __syncthreads();
    for (int off = 128; off > 0; off >>= 1) {
        if (threadIdx.x < off) sred[threadIdx.x] += sred[threadIdx.x + off];
        __syncthreads();
    }
    if (threadIdx.x == 0) out[b] = sred[0];
}

// ---------------------------------------------------------------------------
extern "C" void kernel_launch(void* const* d_in, const int* in_sizes, int n_in,
                              void* d_out, int out_size, void* d_ws, size_t ws_size,
                              hipStream_t stream) {
    const float* ps      = (const float*)d_in[0];   // [N,128]
    const float* sp      = (const float*)d_in[1];   // [4,256,128] unit rows
    const float* w       = (const float*)d_in[2];   // [4,256]
    const int*   species = (const int*)d_in[3];     // [N]
    const int*   sid     = (const int*)d_in[4];     // [N] sorted
    // d_in[5] = n_structures (device scalar); grid size comes from out_size.

    int N        = in_sizes[0] / FEAT;
    int n_struct = out_size;

    char* ws = (char*)d_ws;
    float* Aall    = (float*)ws;                       // 4*128*128 f32 = 262144 B
    int*   counts  = (int*)(ws + 262144);
    int*   base    = (int*)(ws + 262176);              // [5]
    int*   tileOff = (int*)(ws + 262208);              // [5]
    int*   cursor  = (int*)(ws + 262240);
    int*   idxlist = (int*)(ws + 262272);              // [N + 64]
    size_t idx_end = 262272 + (size_t)(N + 64) * sizeof(int);
    size_t pa_off  = (idx_end + 255) & ~(size_t)255;
    float* per_atom = (float*)(ws + pa_off);           // [N]

    int n_pad = N + 64;
    k_init   <<<(n_pad + 255) / 256, 256, 0, stream>>>(counts, cursor, idxlist, n_pad);
    k_build_A<<<NSPEC * FEAT, FEAT, 0, stream>>>(sp, w, Aall);
    k_count  <<<(N + 255) / 256, 256, 0, stream>>>(species, N, counts);
    k_scan   <<<1, 32, 0, stream>>>(counts, base, tileOff);
    k_scatter<<<(N + 255) / 256, 256, 0, stream>>>(species, N, base, cursor, idxlist);

    int maxTiles = (N + 15) / 16 + NSPEC;              // upper bound incl. padding
    k_main   <<<maxTiles, 32, 0, stream>>>(ps, Aall, idxlist, base, tileOff, per_atom);

    k_segsum <<<n_struct, 256, 0, stream>>>(per_atom, sid, N, (float*)d_out);
}